// GAT_Net_48206712930324
// MI455X (gfx1250) — compile-verified
//
#include <hip/hip_runtime.h>
#include <math.h>

#define N_NODES 50000
#define E_EDGES 800000
#define E_TOT   (E_EDGES + N_NODES)   // 850000, GATConv adds self-loops
#define F_INF   128
#define HIDF    32
#define HEADS   4
#define C_OUT   40
#define NEG_SLOPE 0.2f

typedef float v2f __attribute__((ext_vector_type(2)));
typedef float v8f __attribute__((ext_vector_type(8)));

// ---------------------------------------------------------------------------
// fp32 WMMA GEMM:  D[M x NC] = A[M x K] * B[K x NC], row-major.
// One 16x16 output tile per wave32; wave w in the block owns column tile w.
// grid.x = M/16 tiles, blockDim.x = 32 * numColTiles.
// ISA layouts (CDNA5 ISA §7.12.2):
//   A 16x4 : M = lane%16 ; VGPR0 holds K = k+2*(lane/16), VGPR1 = that+1
//   B 4x16 : N = lane%16 ; same K split per lane half
//   C/D    : row = r + 8*(lane/16), col = lane%16
// Ragged NC (e.g. 40) handled branch-free: clamp the column and multiply the
// B fragment by a per-lane {0,1} mask hoisted out of the K-loop, so the inner
// loop is a pure load/wmma clause with no exec-mask manipulation.
// ---------------------------------------------------------------------------
__global__ void k_gemm_wmma_f32(const float* __restrict__ A,
                                const float* __restrict__ B,
                                float* __restrict__ D,
                                int M, int K, int NC) {
  const int tileM = blockIdx.x;
  const int wave  = threadIdx.x >> 5;
  const int lane  = threadIdx.x & 31;
  const int half  = lane >> 4;       // 0 or 1
  const int l16   = lane & 15;
  const int col   = wave * 16 + l16;
  const int row   = tileM * 16 + l16;

  const bool  colOk = col < NC;
  const int   colC  = colOk ? col : (NC - 1);   // safe address for dead lanes
  const float bmask = colOk ? 1.f : 0.f;        // zero dead columns' B operand

  const float* __restrict__ arow = A + (size_t)row * K + 2 * half;
  const float* __restrict__ bcol = B + (size_t)(2 * half) * NC + colC;

  v8f acc = {};
  for (int k = 0; k < K; k += 4) {
    v2f a, b;
    a.x = arow[k];
    a.y = arow[k + 1];
    b.x = bcol[(size_t)k * NC] * bmask;
    b.y = bcol[(size_t)(k + 1) * NC] * bmask;
    acc = __builtin_amdgcn_wmma_f32_16x16x4_f32(
        /*neg_a=*/false, a, /*neg_b=*/false, b,
        /*c_mod=*/(short)0, acc, /*reuse_a=*/false, /*reuse_b=*/false);
  }
  if (colOk) {
#pragma unroll
    for (int r = 0; r < 8; ++r) {
      const int rr = tileM * 16 + r + 8 * half;
      D[(size_t)rr * NC + col] = acc[r];
    }
  }
}

// ---------------------------------------------------------------------------
// helpers
// ---------------------------------------------------------------------------
__device__ __forceinline__ void atomicMaxF(float* addr, float val) {
  unsigned int* ua = reinterpret_cast<unsigned int*>(addr);
  unsigned int old = __float_as_uint(*addr);
  while (__uint_as_float(old) < val) {
    unsigned int assumed = old;
    old = atomicCAS(ua, assumed, __float_as_uint(val));
    if (old == assumed) break;
  }
}

__device__ __forceinline__ void edgeSD(const int* __restrict__ ei, int e,
                                       int& s, int& d) {
  if (e < E_EDGES) { s = ei[e]; d = ei[E_EDGES + e]; }
  else             { s = e - E_EDGES; d = s; }
}

__global__ void k_fill(float* __restrict__ p, size_t n, float v) {
  size_t i = (size_t)blockIdx.x * blockDim.x + threadIdx.x;
  if (i < n) p[i] = v;
}

// ---------------------------------------------------------------------------
// attention dot products, layer 1: one wave per node; lane = feature in head
// e_src[n,h] = sum_f h1[n, h*32+f] * att_src[h,f]
// ---------------------------------------------------------------------------
__global__ void k_att1(const float* __restrict__ h1,
                       const float* __restrict__ asrc,
                       const float* __restrict__ adst,
                       float* __restrict__ esrc, float* __restrict__ edst) {
  const int n    = blockIdx.x * (blockDim.x >> 5) + (threadIdx.x >> 5);
  const int lane = threadIdx.x & 31;
  if (n >= N_NODES) return;
  const float* row = h1 + (size_t)n * (HEADS * HIDF);
  float vs[HEADS], vd[HEADS];
#pragma unroll
  for (int h = 0; h < HEADS; ++h) {
    const float x = row[h * HIDF + lane];
    vs[h] = x * asrc[h * HIDF + lane];
    vd[h] = x * adst[h * HIDF + lane];
  }
#pragma unroll
  for (int h = 0; h < HEADS; ++h)
    for (int off = 16; off; off >>= 1) {
      vs[h] += __shfl_xor(vs[h], off, 32);
      vd[h] += __shfl_xor(vd[h], off, 32);
    }
  if (lane == 0) {
#pragma unroll
    for (int h = 0; h < HEADS; ++h) {
      esrc[n * HEADS + h] = vs[h];
      edst[n * HEADS + h] = vd[h];
    }
  }
}

// attention dots, layer 2 (H=1, C=40): one wave per node
__global__ void k_att2(const float* __restrict__ h2,
                       const float* __restrict__ asrc,
                       const float* __restrict__ adst,
                       float* __restrict__ esrc, float* __restrict__ edst) {
  const int n    = blockIdx.x * (blockDim.x >> 5) + (threadIdx.x >> 5);
  const int lane = threadIdx.x & 31;
  if (n >= N_NODES) return;
  const float* row = h2 + (size_t)n * C_OUT;
  float vs = 0.f, vd = 0.f;
  for (int c = lane; c < C_OUT; c += 32) {
    const float x = row[c];
    vs += x * asrc[c];
    vd += x * adst[c];
  }
  for (int off = 16; off; off >>= 1) {
    vs += __shfl_xor(vs, off, 32);
    vd += __shfl_xor(vd, off, 32);
  }
  if (lane == 0) { esrc[n] = vs; edst[n] = vd; }
}

// ---------------------------------------------------------------------------
// edge pass A: alpha = leaky_relu(e_src[s]+e_dst[d]); atomic segment max
// ---------------------------------------------------------------------------
__global__ void k_edge_alpha(const int* __restrict__ ei,
                             const float* __restrict__ esrc,
                             const float* __restrict__ edst,
                             float* __restrict__ alpha,
                             float* __restrict__ amax, int H) {
  const int e = blockIdx.x * blockDim.x + threadIdx.x;
  if (e >= E_TOT) return;
  int s, d; edgeSD(ei, e, s, d);
  for (int h = 0; h < H; ++h) {
    float a = esrc[s * H + h] + edst[d * H + h];
    a = a > 0.f ? a : NEG_SLOPE * a;
    alpha[(size_t)e * H + h] = a;
    atomicMaxF(&amax[d * H + h], a);
  }
}

// edge pass B: ex = exp(alpha - amax[d]); overwrite alpha; atomic denom
__global__ void k_edge_exp(const int* __restrict__ ei,
                           float* __restrict__ alpha,
                           const float* __restrict__ amax,
                           float* __restrict__ denom, int H) {
  const int e = blockIdx.x * blockDim.x + threadIdx.x;
  if (e >= E_TOT) return;
  int s, d; edgeSD(ei, e, s, d);
  for (int h = 0; h < H; ++h) {
    const float ex = expf(alpha[(size_t)e * H + h] - amax[d * H + h]);
    alpha[(size_t)e * H + h] = ex;
    atomicAdd(&denom[d * H + h], ex);
  }
}

// edge pass C layer1: wave per edge; lane = feature within head
__global__ void k_aggr1(const int* __restrict__ ei,
                        const float* __restrict__ h1,
                        const float* __restrict__ ex,
                        const float* __restrict__ denom,
                        float* __restrict__ agg) {
  const int e    = blockIdx.x * (blockDim.x >> 5) + (threadIdx.x >> 5);
  const int lane = threadIdx.x & 31;
  if (e >= E_TOT) return;
  int s, d; edgeSD(ei, e, s, d);
#pragma unroll
  for (int h = 0; h < HEADS; ++h) {
    const float coef = ex[(size_t)e * HEADS + h] / (denom[d * HEADS + h] + 1e-16f);
    const float v = h1[(size_t)s * (HEADS * HIDF) + h * HIDF + lane] * coef;
    atomicAdd(&agg[(size_t)d * (HEADS * HIDF) + h * HIDF + lane], v);
  }
}

// edge pass C layer2 (H=1, 40 features)
__global__ void k_aggr2(const int* __restrict__ ei,
                        const float* __restrict__ h2,
                        const float* __restrict__ ex,
                        const float* __restrict__ denom,
                        float* __restrict__ agg) {
  const int e    = blockIdx.x * (blockDim.x >> 5) + (threadIdx.x >> 5);
  const int lane = threadIdx.x & 31;
  if (e >= E_TOT) return;
  int s, d; edgeSD(ei, e, s, d);
  const float coef = ex[e] / (denom[d] + 1e-16f);
  for (int c = lane; c < C_OUT; c += 32) {
    atomicAdd(&agg[(size_t)d * C_OUT + c], h2[(size_t)s * C_OUT + c] * coef);
  }
}

// h2in = elu(agg1 + bias1)
__global__ void k_elu_bias(const float* __restrict__ agg,
                           const float* __restrict__ bias,
                           float* __restrict__ hout) {
  const size_t i = (size_t)blockIdx.x * blockDim.x + threadIdx.x;
  if (i >= (size_t)N_NODES * (HEADS * HIDF)) return;
  const float v = agg[i] + bias[i & (HEADS * HIDF - 1)];
  hout[i] = v > 0.f ? v : expf(v) - 1.f;
}

// final: out = agg2 + bias2; write log_softmax then softmax. Wave per node.
__global__ void k_final(const float* __restrict__ agg2,
                        const float* __restrict__ bias2,
                        float* __restrict__ out) {
  const int n    = blockIdx.x * (blockDim.x >> 5) + (threadIdx.x >> 5);
  const int lane = threadIdx.x & 31;
  if (n >= N_NODES) return;
  const float* row = agg2 + (size_t)n * C_OUT;
  const bool has2 = (lane + 32) < C_OUT;            // lanes 0..7
  const float a = row[lane] + bias2[lane];          // lane < 32 < 40 always valid
  const float b = has2 ? row[lane + 32] + bias2[lane + 32] : -INFINITY;
  float m = fmaxf(a, b);
  for (int off = 16; off; off >>= 1) m = fmaxf(m, __shfl_xor(m, off, 32));
  const float ea = expf(a - m);
  const float eb = has2 ? expf(b - m) : 0.f;
  float sum = ea + eb;
  for (int off = 16; off; off >>= 1) sum += __shfl_xor(sum, off, 32);
  const float ls = logf(sum);
  const size_t o1 = (size_t)n * C_OUT;
  const size_t o2 = (size_t)N_NODES * C_OUT + o1;
  out[o1 + lane] = (a - m) - ls;
  out[o2 + lane] = ea / sum;
  if (has2) {
    out[o1 + lane + 32] = (b - m) - ls;
    out[o2 + lane + 32] = eb / sum;
  }
}

// ---------------------------------------------------------------------------
extern "C" void kernel_launch(void* const* d_in, const int* in_sizes, int n_in,
                              void* d_out, int out_size, void* d_ws, size_t ws_size,
                              hipStream_t stream) {
  const float* x     = (const float*)d_in[0];
  const int*   ei    = (const int*)d_in[1];
  const float* W1    = (const float*)d_in[2];
  const float* asrc1 = (const float*)d_in[3];
  const float* adst1 = (const float*)d_in[4];
  const float* bias1 = (const float*)d_in[5];
  const float* W2    = (const float*)d_in[6];
  const float* asrc2 = (const float*)d_in[7];
  const float* adst2 = (const float*)d_in[8];
  const float* bias2 = (const float*)d_in[9];
  float* out = (float*)d_out;

  // workspace layout (floats)
  float* W = (float*)d_ws;
  size_t o = 0;
  float* h1     = W + o; o += (size_t)N_NODES * 128;   // also reused as h2in
  float* agg1   = W + o; o += (size_t)N_NODES * 128;
  float* esrc1  = W + o; o += (size_t)N_NODES * HEADS;
  float* edst1  = W + o; o += (size_t)N_NODES * HEADS;
  float* amax1  = W + o; o += (size_t)N_NODES * HEADS;
  float* den1   = W + o; o += (size_t)N_NODES * HEADS;
  float* alpha1 = W + o; o += (size_t)E_TOT * HEADS;
  float* h2     = W + o; o += (size_t)N_NODES * C_OUT;
  float* esrc2  = W + o; o += (size_t)N_NODES;
  float* edst2  = W + o; o += (size_t)N_NODES;
  float* amax2  = W + o; o += (size_t)N_NODES;
  float* den2   = W + o; o += (size_t)N_NODES;
  float* alpha2 = W + o; o += (size_t)E_TOT;
  float* agg2   = W + o; o += (size_t)N_NODES * C_OUT;

  const int TPB = 256;
  const int nodeWaveBlocks = (N_NODES + 7) / 8;   // 8 waves / 256-thread block
  const int edgeBlocks     = (E_TOT + TPB - 1) / TPB;
  const int edgeWaveBlocks = (E_TOT + 7) / 8;
  const int mTiles         = N_NODES / 16;        // 50000/16 = 3125 exactly

  // ---- init ----
  hipMemsetAsync(agg1, 0, (size_t)N_NODES * 128 * sizeof(float), stream);
  hipMemsetAsync(den1, 0, (size_t)N_NODES * HEADS * sizeof(float), stream);
  hipMemsetAsync(agg2, 0, (size_t)N_NODES * C_OUT * sizeof(float), stream);
  hipMemsetAsync(den2, 0, (size_t)N_NODES * sizeof(float), stream);
  {
    size_t n1 = (size_t)N_NODES * HEADS;
    k_fill<<<(int)((n1 + TPB - 1) / TPB), TPB, 0, stream>>>(amax1, n1, -INFINITY);
    size_t n2 = (size_t)N_NODES;
    k_fill<<<(int)((n2 + TPB - 1) / TPB), TPB, 0, stream>>>(amax2, n2, -INFINITY);
  }

  // ---- layer 1 ----
  k_gemm_wmma_f32<<<mTiles, 8 * 32, 0, stream>>>(x, W1, h1, N_NODES, 128, 128);
  k_att1<<<nodeWaveBlocks, TPB, 0, stream>>>(h1, asrc1, adst1, esrc1, edst1);
  k_edge_alpha<<<edgeBlocks, TPB, 0, stream>>>(ei, esrc1, edst1, alpha1, amax1, HEADS);
  k_edge_exp<<<edgeBlocks, TPB, 0, stream>>>(ei, alpha1, amax1, den1, HEADS);
  k_aggr1<<<edgeWaveBlocks, TPB, 0, stream>>>(ei, h1, alpha1, den1, agg1);

  // h2in = elu(agg1 + bias1), overwriting h1 (dead after k_aggr1)
  {
    size_t n = (size_t)N_NODES * 128;
    k_elu_bias<<<(int)((n + TPB - 1) / TPB), TPB, 0, stream>>>(agg1, bias1, h1);
  }

  // ---- layer 2 ----
  k_gemm_wmma_f32<<<mTiles, 3 * 32, 0, stream>>>(h1, W2, h2, N_NODES, 128, C_OUT);
  k_att2<<<nodeWaveBlocks, TPB, 0, stream>>>(h2, asrc2, adst2, esrc2, edst2);
  k_edge_alpha<<<edgeBlocks, TPB, 0, stream>>>(ei, esrc2, edst2, alpha2, amax2, 1);
  k_edge_exp<<<edgeBlocks, TPB, 0, stream>>>(ei, alpha2, amax2, den2, 1);
  k_aggr2<<<edgeWaveBlocks, TPB, 0, stream>>>(ei, h2, alpha2, den2, agg2);

  // ---- log_softmax / softmax ----
  k_final<<<nodeWaveBlocks, TPB, 0, stream>>>(agg2, bias2, out);
}